// PointPillarScatter_12824772346245
// MI455X (gfx1250) — compile-verified
//
#include <hip/hip_runtime.h>
#include <hip/hip_bf16.h>

// ---------------------------------------------------------------------------
// PointPillarScatter for MI455X (gfx1250).
// - Bulk constant fills (~97% of bytes) run on the Tensor Data Mover: one
//   iterated tensor_store_from_lds covers up to 256 x 64KB = 16MB per op.
// - Deterministic scatter: highest point index wins per cell (winner pass),
//   matching sequential last-update-wins semantics.
// - 2x2 maxpool fused into the cen scatter via exact float atomic-max + a
//   per-region occupancy bitmask (no 944MB 960x960 intermediates).
// ---------------------------------------------------------------------------

#define HW480 (480LL * 480LL)          // 230400 cells per (b, c) plane
#define HW960 (960LL * 960LL)          // 921600 cells per (b) cen plane
#define LOUT  (2LL * 64LL * HW480)     // one [B,64,480,480] output = 29,491,200 floats
#define TDM_CHUNK 16384                // elements per TDM tile (64 KB LDS)
#define TDM_MAX_ITERS 256              // descriptor iteration limit

typedef __attribute__((ext_vector_type(4))) unsigned int uint32x4_t_;
typedef __attribute__((ext_vector_type(8))) int          int32x8_t_;
typedef __attribute__((ext_vector_type(4))) int          int32x4_t_;

// ---------------------------------------------------------------------------
// Issue one TDM store: 1-row tile of `cnt` 4B elements from LDS, replayed
// `iters` times with global_addr_increment = TDM_CHUNK elements and
// lds_addr_increment = 0 (same LDS tile every iteration).
// D# bit layout per CDNA5 ISA ch.8.
// ---------------------------------------------------------------------------
__device__ __forceinline__ void tdm_store_1row(unsigned long long ga, unsigned lds,
                                               unsigned cnt, unsigned iters) {
    uint32x4_t_ g0;
    g0[0] = 1u;                                    // count=1 (valid), no gather
    g0[1] = lds;                                   // lds_addr        [63:32]
    g0[2] = (unsigned)(ga & 0xFFFFFFFFu);          // global_addr     [95:64]
    g0[3] = (unsigned)((ga >> 32) & 0x01FFFFFFu)   // global_addr     [120:96]
          | (2u << 30);                            // type=2 ("image")[127:126]

    int32x8_t_ g1;
    g1[0] = (int)((2u << 16)                       // data_size=4B
          | ((iters > 1u) ? (1u << 19) : 0u));     // iterate_enable
    g1[1] = (int)((cnt & 0xFFFFu) << 16);          // tensor_dim0 lo16   @ 63:48
    g1[2] = (int)(((cnt >> 16) & 0xFFFFu)          // tensor_dim0 hi16   @ 79:64
          | (1u << 16));                           // tensor_dim1=1 lo16 @ 95:80
    g1[3] = (int)((cnt & 0xFFFFu) << 16);          // tile_dim0=cnt      @ 127:112
    g1[4] = 1;                                     // tile_dim1=1, tile_dim2=0
    g1[5] = (int)cnt;                              // tensor_dim0_stride lo32
    g1[6] = 0;
    g1[7] = 0;

    int32x4_t_ g2;                                 // iterate-mode group 2
    g2[0] = 0;                                     // tensor_dim2 unused
    g2[1] = 0;                                     // lds_addr_increment = 0
    g2[2] = (int)(unsigned)TDM_CHUNK;              // global_addr_increment lo32 (elems)
    g2[3] = (int)(((iters - 1u) & 0xFFFFu) << 16); // inc hi16=0, iterate_count @ 127:112

    int32x4_t_ g3  = {0, 0, 0, 0};
    int32x8_t_ gz8 = {0, 0, 0, 0, 0, 0, 0, 0};     // extra group (clang-23 arity)

    __builtin_amdgcn_tensor_store_from_lds(g0, g1, g2, g3, gz8, 0);
}

// ---------------------------------------------------------------------------
// TDM constant fill of `n` 4B elements with bit pattern `bits`.
// Each block owns up to 256 chunks of 16384 elements: one iterated DMA for
// the full chunks, one extra DMA for a partial tail chunk.
// ---------------------------------------------------------------------------
__global__ void fill_tdm_kernel(unsigned int* __restrict__ dst, long long n,
                                unsigned int bits) {
    __shared__ unsigned int lbuf[TDM_CHUNK];
    const long long chunks = (n + TDM_CHUNK - 1) / TDM_CHUNK;
    const long long first  = (long long)blockIdx.x * TDM_MAX_ITERS;
    long long myChunks = chunks - first;
    if (myChunks > TDM_MAX_ITERS) myChunks = TDM_MAX_ITERS;

    uint4* lb4 = (uint4*)lbuf;
    const uint4 v4 = make_uint4(bits, bits, bits, bits);
    for (int i = threadIdx.x; i < TDM_CHUNK / 4; i += blockDim.x) lb4[i] = v4;
    __syncthreads();

    if (threadIdx.x == 0) {
        const unsigned  lds      = (unsigned)(uintptr_t)&lbuf[0];
        const long long lastElem = (first + myChunks) * TDM_CHUNK;   // exclusive
        const bool      hasTail  = lastElem > n;                     // partial last chunk
        const long long fullIt   = hasTail ? myChunks - 1 : myChunks;

        if (fullIt > 0)
            tdm_store_1row((unsigned long long)(uintptr_t)(dst + first * TDM_CHUNK),
                           lds, TDM_CHUNK, (unsigned)fullIt);
        if (hasTail) {
            const long long tailStart = (first + fullIt) * TDM_CHUNK;
            tdm_store_1row((unsigned long long)(uintptr_t)(dst + tailStart),
                           lds, (unsigned)(n - tailStart), 1u);
        }
        __builtin_amdgcn_s_wait_tensorcnt(0);
    }
}

// Exact IEEE float atomic max via integer atomics:
//  v >= 0: signed-int max; v < 0: unsigned-int min (keeps any non-negative).
__device__ __forceinline__ void atomicMaxFloat(float* addr, float v) {
    if (v >= 0.0f) atomicMax((int*)addr, __float_as_int(v));
    else           atomicMin((unsigned int*)addr, __float_as_uint(v));
}

// ---------------------------------------------------------------------------
// Winner pass: highest point index wins each cell (deterministic last-wins).
// win is pre-filled with -1. planeSize = ny*nx, grid pitch = nx.
// ---------------------------------------------------------------------------
__global__ void winner_kernel(const int* __restrict__ coords,
                              int* __restrict__ win,
                              long long planeSize, int nx, int P) {
    const int p = blockIdx.x * blockDim.x + threadIdx.x;
    if (p >= P) return;
    const int b = coords[p * 4 + 0];
    const int y = coords[p * 4 + 2];
    const int x = coords[p * 4 + 3];
    atomicMax(&win[(long long)b * planeSize + (long long)y * nx + x], p);
}

// ---------------------------------------------------------------------------
// Direct scatter (480 grids): 1 block = 1 point, 64 threads = 64 channels.
// Only the winning point of a cell writes. out2 pre-offset to slice ch0, b0.
// ---------------------------------------------------------------------------
__global__ void scatter_direct_kernel(const float* __restrict__ feat,
                                      const int*   __restrict__ coords,
                                      const int*   __restrict__ win,
                                      float* __restrict__ out1,   // [B,64,480,480]
                                      float* __restrict__ out2) { // cat slice base
    const int p = blockIdx.x;
    const int c = threadIdx.x;
    const int b = coords[p * 4 + 0];
    const int y = coords[p * 4 + 2];
    const int x = coords[p * 4 + 3];
    const long long cell = (long long)y * 480 + x;
    if (win[(long long)b * HW480 + cell] != p) return;
    const float v = feat[p * 64 + c];
    out1[((long long)b * 64  + c) * HW480 + cell] = v;
    out2[((long long)b * 256 + c) * HW480 + cell] = v;
}

// ---------------------------------------------------------------------------
// Cen scatter + fused 2x2 maxpool: winners of the 960 grid atomic-max into
// the pooled cat slice (pre-init -FLT_MAX) and OR their quadrant bit.
// ---------------------------------------------------------------------------
__global__ void scatter_cen_kernel(const float* __restrict__ feat,
                                   const int*   __restrict__ coords,
                                   const int*   __restrict__ win960,
                                   float*        __restrict__ catSlice,
                                   unsigned int* __restrict__ mask) {   // [B,480,480]
    const int p = blockIdx.x;
    const int c = threadIdx.x;
    const int b = coords[p * 4 + 0];
    const int y = coords[p * 4 + 2];
    const int x = coords[p * 4 + 3];
    if (win960[(long long)b * HW960 + (long long)y * 960 + x] != p) return;
    const int py = y >> 1, px = x >> 1;
    const long long cell = (long long)py * 480 + px;
    atomicMaxFloat(&catSlice[((long long)b * 256 + c) * HW480 + cell],
                   feat[p * 64 + c]);
    if (c == 0)
        atomicOr(&mask[(long long)b * HW480 + cell], 1u << (((y & 1) << 1) | (x & 1)));
}

// ---------------------------------------------------------------------------
// Fix-up: empty region -> 0; fully-occupied -> raw max; else max(raw, 0)
// (empty source cells of the 2x2 window contribute the grid's zero init).
// ---------------------------------------------------------------------------
__global__ void fixup_cen_kernel(float* __restrict__ catSlice,
                                 const unsigned int* __restrict__ mask) {
    const long long cellFlat = blockIdx.x;            // [0, B*HW480)
    const int       b  = (int)(cellFlat / HW480);
    const long long lc = cellFlat % HW480;
    const unsigned  m  = mask[cellFlat];
    const int       c  = threadIdx.x;
    const long long idx = ((long long)b * 256 + c) * HW480 + lc;
    const float v = catSlice[idx];
    catSlice[idx] = (m == 0u) ? 0.0f : ((m == 0xFu) ? v : fmaxf(v, 0.0f));
}

extern "C" void kernel_launch(void* const* d_in, const int* in_sizes, int n_in,
                              void* d_out, int out_size, void* d_ws, size_t ws_size,
                              hipStream_t stream) {
    const float* radar_cen_feat   = (const float*)d_in[0];
    const int*   radar_cen_coords = (const int*)  d_in[1];
    const float* lidar_cen_feat   = (const float*)d_in[2];
    const int*   lidar_cen_coords = (const int*)  d_in[3];
    const float* lidar_feat       = (const float*)d_in[4];
    const int*   lidar_coords     = (const int*)  d_in[5];
    const float* radar_feat       = (const float*)d_in[6];
    const int*   radar_coords     = (const int*)  d_in[7];

    const int P_rcen = in_sizes[0] / 64;
    const int P_lcen = in_sizes[2] / 64;
    const int P_lid  = in_sizes[4] / 64;
    const int P_rad  = in_sizes[6] / 64;

    float* out = (float*)d_out;
    float* cat = out + 2LL * LOUT;                     // [B,256,480,480]

    // Workspace layout (4B elements).
    unsigned int* ws     = (unsigned int*)d_ws;
    unsigned int* maskL  = ws;                               // 2*HW480
    unsigned int* maskR  = maskL  + 2LL * HW480;             // 2*HW480
    int*          winL   = (int*)(maskR + 2LL * HW480);      // 2*HW480
    int*          winR   = winL   + 2LL * HW480;             // 2*HW480
    int*          winLC  = winR   + 2LL * HW480;             // 2*HW960
    int*          winRC  = winLC  + 2LL * HW960;             // 2*HW960

    const long long SLICE = 64LL * HW480;              // one 64-ch slice per batch
    const unsigned  ZBITS = 0x00000000u;               // 0.0f / mask 0
    const unsigned  NBITS = 0xFF7FFFFFu;               // -FLT_MAX bit pattern
    const unsigned  WBITS = 0xFFFFFFFFu;               // winner init = -1

    auto fill = [&](void* p, long long n, unsigned bits) {
        long long chunks = (n + TDM_CHUNK - 1) / TDM_CHUNK;
        int blocks = (int)((chunks + TDM_MAX_ITERS - 1) / TDM_MAX_ITERS);
        fill_tdm_kernel<<<dim3(blocks), dim3(256), 0, stream>>>(
            (unsigned int*)p, n, bits);
    };

    // --- TDM constant fills -------------------------------------------------
    fill(out, 2LL * LOUT + SLICE, ZBITS);              // lidar+radar out, cat b0 lidar
    fill(cat + 1 * SLICE, SLICE, NBITS);               // b0 lidar_cen
    fill(cat + 2 * SLICE, SLICE, ZBITS);               // b0 radar
    fill(cat + 3 * SLICE, SLICE, NBITS);               // b0 radar_cen
    fill(cat + 4 * SLICE, SLICE, ZBITS);               // b1 lidar
    fill(cat + 5 * SLICE, SLICE, NBITS);               // b1 lidar_cen
    fill(cat + 6 * SLICE, SLICE, ZBITS);               // b1 radar
    fill(cat + 7 * SLICE, SLICE, NBITS);               // b1 radar_cen
    fill(maskL, 4LL * HW480, ZBITS);                   // occupancy masks -> 0
    fill(winL, 4LL * HW480 + 4LL * HW960, WBITS);      // all winner grids -> -1

    // --- winner passes (deterministic duplicate resolution) ------------------
    winner_kernel<<<dim3((P_lid  + 255) / 256), dim3(256), 0, stream>>>(
        lidar_coords, winL, HW480, 480, P_lid);
    winner_kernel<<<dim3((P_rad  + 255) / 256), dim3(256), 0, stream>>>(
        radar_coords, winR, HW480, 480, P_rad);
    winner_kernel<<<dim3((P_lcen + 255) / 256), dim3(256), 0, stream>>>(
        lidar_cen_coords, winLC, HW960, 960, P_lcen);
    winner_kernel<<<dim3((P_rcen + 255) / 256), dim3(256), 0, stream>>>(
        radar_cen_coords, winRC, HW960, 960, P_rcen);

    // --- scatters ------------------------------------------------------------
    scatter_direct_kernel<<<dim3(P_lid), dim3(64), 0, stream>>>(
        lidar_feat, lidar_coords, winL, out, cat /* ch 0 */);
    scatter_direct_kernel<<<dim3(P_rad), dim3(64), 0, stream>>>(
        radar_feat, radar_coords, winR, out + LOUT, cat + 128 * HW480 /* ch 128 */);

    scatter_cen_kernel<<<dim3(P_lcen), dim3(64), 0, stream>>>(
        lidar_cen_feat, lidar_cen_coords, winLC, cat + 64 * HW480 /* ch 64 */, maskL);
    scatter_cen_kernel<<<dim3(P_rcen), dim3(64), 0, stream>>>(
        radar_cen_feat, radar_cen_coords, winRC, cat + 192 * HW480 /* ch 192 */, maskR);

    // --- maxpool fix-up ------------------------------------------------------
    fixup_cen_kernel<<<dim3((unsigned)(2LL * HW480)), dim3(64), 0, stream>>>(
        cat + 64 * HW480, maskL);
    fixup_cen_kernel<<<dim3((unsigned)(2LL * HW480)), dim3(64), 0, stream>>>(
        cat + 192 * HW480, maskR);
}